// HebbianLayer_49082886258997
// MI455X (gfx1250) — compile-verified
//
#include <hip/hip_runtime.h>

#define IN_F   1024
#define OUT_F  2048
#define BATCH  8192
#define LR     0.02f
#define ANTI   0.4f
#define PREC   1e-30f
#define NBX    4            // N-blocks in GEMM grid (2048/512)

typedef __attribute__((ext_vector_type(2))) float v2f;
typedef __attribute__((ext_vector_type(8))) float v8f;

// ---------------------------------------------------------------------------
// Async global->LDS staging (CDNA5 GLOBAL_LOAD_ASYNC_TO_LDS_B128, ASYNCcnt),
// with graceful fallback to direct LDS stores if the builtin is absent.
// Builtin signature (from clang diagnostic): (v4i AS1*, v4i AS3*, Ii, Ii).
// ---------------------------------------------------------------------------
#if __has_builtin(__builtin_amdgcn_global_load_async_to_lds_b128)
#define HAVE_ASYNC 1
typedef int v4i_vs __attribute__((vector_size(16)));
typedef __attribute__((address_space(1))) v4i_vs* as1_v4i_ptr;
typedef __attribute__((address_space(3))) v4i_vs* as3_v4i_ptr;
#else
#define HAVE_ASYNC 0
#endif

__device__ __forceinline__ void stage16(const float* g, float* l) {
#if HAVE_ASYNC
  __builtin_amdgcn_global_load_async_to_lds_b128(
      (as1_v4i_ptr)(void*)g, (as3_v4i_ptr)(void*)l, 0, 0);
#else
  *(float4*)l = *(const float4*)g;
#endif
}

__device__ __forceinline__ void stage_wait() {
#if HAVE_ASYNC
#if __has_builtin(__builtin_amdgcn_s_wait_asynccnt)
  __builtin_amdgcn_s_wait_asynccnt(0);
#else
  asm volatile("s_wait_asynccnt 0" ::: "memory");
#endif
#endif
}

__device__ __forceinline__ void top2_merge(float& v0, int& i0, float& v1, int& i1,
                                           float v, int i) {
  if (v > v0)      { v1 = v0; i1 = i0; v0 = v; i0 = i; }
  else if (v > v1) { v1 = v;  i1 = i; }
}

// ---------------------------------------------------------------------------
// Kernel 1: y = x @ w^T via V_WMMA_F32_16X16X4_F32, fused partial top-2.
// Block: 8 waves, tile 32(M) x 512(N); wave tile 32 x 64 (8 accumulators).
// A staged in LDS (32 rows x 32 K, stride 36 floats: conflict-free b64 reads).
// A-frag (16x4 f32): lanes 0-15 K={0,1}, lanes 16-31 K={2,3}; M = lane%16.
// B-frag (4x16 f32): B[k][n]=w[n][k]; VGPR0 K=0/2, VGPR1 K=1/3 per half-wave.
// C/D (16x16 f32): VGPR v -> M = v + 8*(lane>=16), N = lane%16.
// Epilogue: per-row top-2 over this block's 512 cols -> ws partial arrays.
// ---------------------------------------------------------------------------
__global__ __launch_bounds__(256) void hebb_gemm_wmma(
    const float* __restrict__ x, const float* __restrict__ w,
    float* __restrict__ y,
    float* __restrict__ pv0, int* __restrict__ pi0,
    float* __restrict__ pv1, int* __restrict__ pi1) {
  __shared__ float s_x[32 * 36];       // padded stride 36
  __shared__ float s_cv[32 * 8 * 2];
  __shared__ int   s_ci[32 * 8 * 2];

  const int tid  = threadIdx.x;
  const int lane = tid & 31;
  const int wave = tid >> 5;
  const int half = lane >> 4;
  const int l16  = lane & 15;

  const int m0 = blockIdx.y * 32;
  const int n0 = blockIdx.x * 512 + wave * 64;

  const float* wrow = w + (size_t)(n0 + l16) * IN_F + half * 2;

  // staging: 256 threads cover 32 rows x 32 floats (16B each)
  const int srow = tid >> 3;
  const int sseg = (tid & 7) * 4;
  const float* gsrc = x + (size_t)(m0 + srow) * IN_F + sseg;
  float* ldst = &s_x[srow * 36 + sseg];

  v8f aA0 = {}, aA1 = {}, aA2 = {}, aA3 = {};   // rows m0..m0+15
  v8f aB0 = {}, aB1 = {}, aB2 = {}, aB3 = {};   // rows m0+16..m0+31

  for (int kc = 0; kc < IN_F; kc += 32) {
    __syncthreads();                    // previous chunk fully consumed
    stage16(gsrc + kc, ldst);
    stage_wait();
    __syncthreads();
#pragma unroll
    for (int kk = 0; kk < 32; kk += 4) {
      v2f a0 = *(const v2f*)&s_x[l16 * 36 + kk + half * 2];
      v2f a1 = *(const v2f*)&s_x[(16 + l16) * 36 + kk + half * 2];
      const float* wp = wrow + kc + kk;
      v2f b0 = *(const v2f*)(wp);
      v2f b1 = *(const v2f*)(wp + (size_t)16 * IN_F);
      v2f b2 = *(const v2f*)(wp + (size_t)32 * IN_F);
      v2f b3 = *(const v2f*)(wp + (size_t)48 * IN_F);
      aA0 = __builtin_amdgcn_wmma_f32_16x16x4_f32(false, a0, false, b0, (short)0, aA0, false, false);
      aA1 = __builtin_amdgcn_wmma_f32_16x16x4_f32(false, a0, false, b1, (short)0, aA1, false, false);
      aA2 = __builtin_amdgcn_wmma_f32_16x16x4_f32(false, a0, false, b2, (short)0, aA2, false, false);
      aA3 = __builtin_amdgcn_wmma_f32_16x16x4_f32(false, a0, false, b3, (short)0, aA3, false, false);
      aB0 = __builtin_amdgcn_wmma_f32_16x16x4_f32(false, a1, false, b0, (short)0, aB0, false, false);
      aB1 = __builtin_amdgcn_wmma_f32_16x16x4_f32(false, a1, false, b1, (short)0, aB1, false, false);
      aB2 = __builtin_amdgcn_wmma_f32_16x16x4_f32(false, a1, false, b2, (short)0, aB2, false, false);
      aB3 = __builtin_amdgcn_wmma_f32_16x16x4_f32(false, a1, false, b3, (short)0, aB3, false, false);
    }
  }

  // ---- store y; fold per-row top-2 into epilogue ----
  float* yb0 = y + (size_t)(m0 + half * 8) * OUT_F + n0 + l16;        // g=0
  float* yb1 = yb0 + (size_t)16 * OUT_F;                              // g=1

  auto top2row = [&](float c0, float c1, float c2, float c3, int r) {
    float v0 = -__builtin_inff(), v1 = -__builtin_inff();
    int   i0 = -1, i1 = -1;
    top2_merge(v0, i0, v1, i1, c0, n0 + 0  + l16);
    top2_merge(v0, i0, v1, i1, c1, n0 + 16 + l16);
    top2_merge(v0, i0, v1, i1, c2, n0 + 32 + l16);
    top2_merge(v0, i0, v1, i1, c3, n0 + 48 + l16);
#pragma unroll
    for (int off = 1; off <= 8; off <<= 1) {   // stays within the 16-lane half
      float ov0 = __shfl_xor(v0, off); int oi0 = __shfl_xor(i0, off);
      float ov1 = __shfl_xor(v1, off); int oi1 = __shfl_xor(i1, off);
      top2_merge(v0, i0, v1, i1, ov0, oi0);
      top2_merge(v0, i0, v1, i1, ov1, oi1);
    }
    if (l16 == 0) {
      const int c = (r * 8 + wave) * 2;
      s_cv[c]     = v0; s_ci[c]     = i0;
      s_cv[c + 1] = v1; s_ci[c + 1] = i1;
    }
  };

#pragma unroll
  for (int v = 0; v < 8; ++v) {
    yb0[(size_t)v * OUT_F + 0]  = aA0[v];
    yb0[(size_t)v * OUT_F + 16] = aA1[v];
    yb0[(size_t)v * OUT_F + 32] = aA2[v];
    yb0[(size_t)v * OUT_F + 48] = aA3[v];
    yb1[(size_t)v * OUT_F + 0]  = aB0[v];
    yb1[(size_t)v * OUT_F + 16] = aB1[v];
    yb1[(size_t)v * OUT_F + 32] = aB2[v];
    yb1[(size_t)v * OUT_F + 48] = aB3[v];
    top2row(aA0[v], aA1[v], aA2[v], aA3[v], v + 8 * half);
    top2row(aB0[v], aB1[v], aB2[v], aB3[v], 16 + v + 8 * half);
  }
  __syncthreads();

  if (tid < 32) {                       // thread r merges 8 waves x 2 slots
    float v0 = -__builtin_inff(), v1 = -__builtin_inff();
    int   i0 = -1, i1 = -1;
    const int base = tid * 16;
#pragma unroll
    for (int c = 0; c < 16; ++c)
      top2_merge(v0, i0, v1, i1, s_cv[base + c], s_ci[base + c]);
    const int slot = (m0 + tid) * NBX + blockIdx.x;
    pv0[slot] = v0; pi0[slot] = i0;
    pv1[slot] = v1; pi1[slot] = i1;
  }
}

// ---------------------------------------------------------------------------
// Kernel 0: zero xx[OUT_F] and nc.
// ---------------------------------------------------------------------------
__global__ void hebb_zero(float* __restrict__ xx, unsigned* __restrict__ nc) {
  const int tid = blockIdx.x * blockDim.x + threadIdx.x;
  if (tid < OUT_F) xx[tid] = 0.0f;
  if (tid == 0) *nc = 0u;
}

// ---------------------------------------------------------------------------
// Kernel 2: merge per-block partials -> final top-2 per batch row, xx accum.
// ---------------------------------------------------------------------------
__global__ __launch_bounds__(256) void hebb_top2_final(
    const float* __restrict__ pv0, const int* __restrict__ pi0,
    const float* __restrict__ pv1, const int* __restrict__ pi1,
    float* __restrict__ xx, int* __restrict__ idx0, int* __restrict__ idx1) {
  const int b = blockIdx.x * 256 + threadIdx.x;   // 0..BATCH-1
  float v0 = -__builtin_inff(), v1 = -__builtin_inff();
  int   i0 = -1, i1 = -1;
#pragma unroll
  for (int nb = 0; nb < NBX; ++nb) {
    const int s = b * NBX + nb;
    top2_merge(v0, i0, v1, i1, pv0[s], pi0[s]);
    top2_merge(v0, i0, v1, i1, pv1[s], pi1[s]);
  }
  idx0[b] = i0;
  idx1[b] = i1;
  atomicAdd(&xx[i0], v0);           // yl = +1.0 at top-1
  atomicAdd(&xx[i1], -ANTI * v1);   // yl = -0.4 at rank-2
}

// ---------------------------------------------------------------------------
// Kernel 3: ds = -xx[o] * w   (float4; replaces separate zero + subtract pass)
// ---------------------------------------------------------------------------
__global__ __launch_bounds__(256) void hebb_init_ds(
    const float* __restrict__ w, const float* __restrict__ xx,
    float* __restrict__ ds) {
  const int i4 = blockIdx.x * 256 + threadIdx.x;    // float4 index
  const float s = -xx[i4 >> 8];                     // IN_F/4 = 256 vecs per row
  const float4 wv = ((const float4*)w)[i4];
  float4 o;
  o.x = s * wv.x; o.y = s * wv.y; o.z = s * wv.z; o.w = s * wv.w;
  ((float4*)ds)[i4] = o;
}

// ---------------------------------------------------------------------------
// Kernel 4: ds += yl @ x (sparse scatter, f32 atomics into L2-resident 8 MB).
// ---------------------------------------------------------------------------
__global__ __launch_bounds__(256) void hebb_scatter(
    const float* __restrict__ x, const int* __restrict__ idx0,
    const int* __restrict__ idx1, float* __restrict__ ds) {
  const int b = blockIdx.x;
  const int j0 = idx0[b];
  const int j1 = idx1[b];
  const float* xr = x + (size_t)b * IN_F;
  float* d0 = ds + (size_t)j0 * IN_F;
  float* d1 = ds + (size_t)j1 * IN_F;
  for (int i = threadIdx.x; i < IN_F; i += 256) {
    const float xv = xr[i];
    atomicAdd(d0 + i, xv);
    atomicAdd(d1 + i, -ANTI * xv);
  }
}

// ---------------------------------------------------------------------------
// Kernel 5: nc = max|ds| (wave-reduced atomicMax on float bits; |ds| >= 0).
// ---------------------------------------------------------------------------
__global__ __launch_bounds__(256) void hebb_maxabs(
    const float* __restrict__ ds, unsigned* __restrict__ ncbits) {
  const int i4 = blockIdx.x * 256 + threadIdx.x;
  const float4 d = ((const float4*)ds)[i4];
  float a = fmaxf(fmaxf(__builtin_fabsf(d.x), __builtin_fabsf(d.y)),
                  fmaxf(__builtin_fabsf(d.z), __builtin_fabsf(d.w)));
#pragma unroll
  for (int off = 16; off > 0; off >>= 1)
    a = fmaxf(a, __shfl_xor(a, off));
  if ((threadIdx.x & 31) == 0)
    atomicMax(ncbits, __float_as_uint(a));
}

// ---------------------------------------------------------------------------
// Kernel 6: new_w = w + LR * ds / nc  (in place over ds staging).
// ---------------------------------------------------------------------------
__global__ __launch_bounds__(256) void hebb_update(
    const float* __restrict__ w, float* __restrict__ ds_newW,
    const unsigned* __restrict__ ncbits) {
  const int i4 = blockIdx.x * 256 + threadIdx.x;
  const float nc = fmaxf(__uint_as_float(*ncbits), PREC);
  const float s = LR / nc;
  const float4 wv = ((const float4*)w)[i4];
  float4 d = ((float4*)ds_newW)[i4];
  d.x = wv.x + s * d.x;
  d.y = wv.y + s * d.y;
  d.z = wv.z + s * d.z;
  d.w = wv.w + s * d.w;
  ((float4*)ds_newW)[i4] = d;
}

// ---------------------------------------------------------------------------
extern "C" void kernel_launch(void* const* d_in, const int* in_sizes, int n_in,
                              void* d_out, int out_size, void* d_ws, size_t ws_size,
                              hipStream_t stream) {
  const float* x = (const float*)d_in[0];   // [8192, 1024]
  const float* w = (const float*)d_in[1];   // [2048, 1024]

  float* y  = (float*)d_out;                          // [8192, 2048]
  float* ds = (float*)d_out + (size_t)BATCH * OUT_F;  // [2048, 1024] staging -> new_w

  char* ws = (char*)d_ws;
  float*    xx   = (float*)(ws);                  // 2048 f
  int*      idx0 = (int*)(ws + 8192);             // 8192 i
  int*      idx1 = (int*)(ws + 40960);            // 8192 i
  unsigned* nc   = (unsigned*)(ws + 73728);       // 1 u32
  float*    pv0  = (float*)(ws + 131072);         // 8192*4 f
  int*      pi0  = (int*)(ws + 262144);           // 8192*4 i
  float*    pv1  = (float*)(ws + 393216);         // 8192*4 f
  int*      pi1  = (int*)(ws + 524288);           // 8192*4 i

  hipLaunchKernelGGL(hebb_zero, dim3(8), dim3(256), 0, stream, xx, nc);

  hipLaunchKernelGGL(hebb_gemm_wmma, dim3(OUT_F / 512, BATCH / 32), dim3(256), 0, stream,
                     x, w, y, pv0, pi0, pv1, pi1);

  hipLaunchKernelGGL(hebb_top2_final, dim3(BATCH / 256), dim3(256), 0, stream,
                     pv0, pi0, pv1, pi1, xx, idx0, idx1);

  hipLaunchKernelGGL(hebb_init_ds, dim3((OUT_F * IN_F / 4) / 256), dim3(256), 0, stream,
                     w, xx, ds);

  hipLaunchKernelGGL(hebb_scatter, dim3(BATCH), dim3(256), 0, stream, x, idx0, idx1, ds);

  hipLaunchKernelGGL(hebb_maxabs, dim3((OUT_F * IN_F / 4) / 256), dim3(256), 0, stream,
                     ds, nc);

  hipLaunchKernelGGL(hebb_update, dim3((OUT_F * IN_F / 4) / 256), dim3(256), 0, stream,
                     w, ds, nc);
}